// CRF_51299089383986
// MI455X (gfx1250) — compile-verified
//
#include <hip/hip_runtime.h>
#include <math.h>

typedef __attribute__((ext_vector_type(2))) float v2f;
typedef __attribute__((ext_vector_type(8))) float v8f;

#define B_BATCH 4096
#define LSEQ    64
#define DDIM    128
#define YNUM    26
#define YPAD    32
#define CREG    1000.0f

// workspace layout (float offsets)
#define WS_LOGLIK 0        // [0, 4096)   per-batch loglik
#define WS_REG    4096     // [4096]      0.5*(||W||^2 + ||T||^2)
#define WS_WEFF   4352     // [4352, 8448) W_eff, 128 x 32 (cols >=26 zero)

// ---------------------------------------------------------------------------
// Kernel 0: fold the 5x5 conv into the projection: W_eff = C^T W  (128x26),
// plus the quadratic regularizer. Tiny, one workgroup.
// ---------------------------------------------------------------------------
__global__ void crf_prep(const float* __restrict__ W, const float* __restrict__ T,
                         const float* __restrict__ K, float* __restrict__ ws) {
  const int tid = threadIdx.x;
  for (int idx = tid; idx < DDIM * YPAD; idx += 256) {
    const int d = idx >> 5, y = idx & 31;
    float acc = 0.f;
    if (y < YNUM) {
      const int rp = d >> 3, cp = d & 7;            // d' = rp*8 + cp  (16x8 image)
      for (int u = 0; u < 5; ++u)
        for (int v = 0; v < 5; ++v) {
          const int r = rp - u + 2, c = cp - v + 2;
          if (r >= 0 && r < 16 && c >= 0 && c < 8)
            acc += K[u * 5 + v] * W[(r * 8 + c) * YNUM + y];
        }
    }
    ws[WS_WEFF + idx] = acc;
  }
  float p = 0.f;
  for (int i = tid; i < DDIM * YNUM; i += 256) { const float w = W[i]; p += w * w; }
  for (int i = tid; i < YNUM * YNUM; i += 256) { const float t = T[i]; p += t * t; }
  __shared__ float red[256];
  red[tid] = p;
  __syncthreads();
  for (int s = 128; s > 0; s >>= 1) { if (tid < s) red[tid] += red[tid + s]; __syncthreads(); }
  if (tid == 0) ws[WS_REG] = 0.5f * red[0];
}

// ---------------------------------------------------------------------------
// Kernel 1: one workgroup per batch element.
//   Phase A: stage X_b (64x128 f32) + W_eff + T + labels into LDS
//   Phase B: e = X_b @ W_eff via V_WMMA_F32_16X16X4_F32 (8 tiles, 1 per wave)
//   Phase C: emission/transition scores + 63-step logsumexp forward recursion
// ---------------------------------------------------------------------------
__global__ __launch_bounds__(256) void crf_batch(const float* __restrict__ X,
                                                 const int* __restrict__ labels,
                                                 const float* __restrict__ T,
                                                 float* __restrict__ ws) {
  __shared__ __align__(16) float Xs[LSEQ][DDIM + 4];  // 132 stride: conflict-free A frags
  __shared__ float Wsm[DDIM][YPAD + 1];               // 33 stride: conflict-free B frags
  __shared__ float Es[LSEQ][YPAD + 1];                // emissions e (64x32)
  __shared__ float Ts[YNUM][28];                      // transition matrix
  __shared__ int   labs[LSEQ];
  __shared__ float alph[2][32];                       // ping-pong alpha
  __shared__ float red[LSEQ];

  const int tid = threadIdx.x;
  const int b = blockIdx.x;
  const float* __restrict__ Xb   = X + (size_t)b * (LSEQ * DDIM);
  const float* __restrict__ weff = ws + WS_WEFF;

  // ---- Phase A: staging (X_b read once from HBM, float4-coalesced) ----
  for (int i = tid; i < (LSEQ * DDIM) / 4; i += 256) {
    const float4 v = ((const float4*)Xb)[i];
    const int row = i >> 5;            // (i*4) / 128
    const int col = (i & 31) << 2;
    *(float4*)&Xs[row][col] = v;       // row stride 528B: 16B aligned
  }
  for (int i = tid; i < DDIM * YPAD; i += 256) Wsm[i >> 5][i & 31] = weff[i];
  for (int i = tid; i < YNUM * YNUM; i += 256) Ts[i / YNUM][i % YNUM] = T[i];
  if (tid < LSEQ) labs[tid] = labels[(size_t)b * LSEQ + tid];
  __syncthreads();

  // ---- Phase B: e = X_b @ W_eff, fp32 WMMA 16x16x4, K=128 in 32 steps ----
  {
    const int wave = tid >> 5, lane = tid & 31;
    const int m16 = (wave >> 1) << 4;      // 4 M tiles
    const int n16 = (wave & 1) << 4;       // 2 N tiles
    const int lh = lane >> 4, li = lane & 15;
    const int row = m16 + li, ncol = n16 + li;
    v8f acc = {0.f, 0.f, 0.f, 0.f, 0.f, 0.f, 0.f, 0.f};
    for (int k = 0; k < DDIM; k += 4) {
      // A 16x4: lanes 0-15 hold K=0/1 in v0/v1, lanes 16-31 hold K=2/3
      v2f a, bb;
      a[0] = Xs[row][k + 2 * lh];
      a[1] = Xs[row][k + 2 * lh + 1];
      // B 4x16: N across lanes within a VGPR, K split across lane halves
      bb[0] = Wsm[k + 2 * lh][ncol];
      bb[1] = Wsm[k + 2 * lh + 1][ncol];
      acc = __builtin_amdgcn_wmma_f32_16x16x4_f32(
          /*neg_a=*/false, a, /*neg_b=*/false, bb,
          /*c_mod=*/(short)0, acc, /*reuse_a=*/false, /*reuse_b=*/false);
    }
    // C/D layout: VGPR r -> M = r + 8*(lane>=16); N = lane&15
#pragma unroll
    for (int r = 0; r < 8; ++r)
      Es[m16 + r + 8 * lh][ncol] = acc[r];
  }
  __syncthreads();

  // ---- Phase C: scores + forward recursion ----
  float em = 0.f, tr = 0.f;
  if (tid < LSEQ) red[tid] = Es[tid][labs[tid]];
  __syncthreads();
  if (tid == 0) { float s = 0.f; for (int i = 0; i < LSEQ; ++i) s += red[i]; em = s; }
  __syncthreads();
  if (tid < LSEQ) red[tid] = (tid < LSEQ - 1) ? Ts[labs[tid]][labs[tid + 1]] : 0.f;
  __syncthreads();
  if (tid == 0) { float s = 0.f; for (int i = 0; i < LSEQ - 1; ++i) s += red[i]; tr = s; }

  if (tid < 32) alph[0][tid] = (tid < YNUM) ? Es[0][tid] : -1e30f;
  __syncthreads();

  int cur = 0;
  for (int t = 1; t < LSEQ; ++t) {
    float nv = 0.f;
    if (tid < YNUM) {
      float m = -3.0e38f;
      for (int i = 0; i < YNUM; ++i) m = fmaxf(m, alph[cur][i] + Ts[i][tid]);
      float s = 0.f;
      for (int i = 0; i < YNUM; ++i) s += __expf(alph[cur][i] + Ts[i][tid] - m);
      nv = Es[t][tid] + m + __logf(s);
    }
    if (tid < YNUM) alph[cur ^ 1][tid] = nv;
    cur ^= 1;
    __syncthreads();
  }

  if (tid == 0) {
    float m = -3.0e38f;
    for (int i = 0; i < YNUM; ++i) m = fmaxf(m, alph[cur][i]);
    float s = 0.f;
    for (int i = 0; i < YNUM; ++i) s += __expf(alph[cur][i] - m);
    const float logZ = m + __logf(s);
    ws[WS_LOGLIK + b] = em + tr - logZ;   // deterministic: no atomics
  }
}

// ---------------------------------------------------------------------------
// Kernel 2: reduce 4096 logliks, apply -C*mean + regularizer.
// ---------------------------------------------------------------------------
__global__ void crf_final(const float* __restrict__ ws, float* __restrict__ out) {
  __shared__ float red[256];
  const int tid = threadIdx.x;
  float s = 0.f;
  for (int i = tid; i < B_BATCH; i += 256) s += ws[WS_LOGLIK + i];
  red[tid] = s;
  __syncthreads();
  for (int st = 128; st > 0; st >>= 1) { if (tid < st) red[tid] += red[tid + st]; __syncthreads(); }
  if (tid == 0) out[0] = -CREG * (red[0] / (float)B_BATCH) + ws[WS_REG];
}

extern "C" void kernel_launch(void* const* d_in, const int* in_sizes, int n_in,
                              void* d_out, int out_size, void* d_ws, size_t ws_size,
                              hipStream_t stream) {
  const float* X      = (const float*)d_in[0];
  const int*   labels = (const int*)d_in[1];
  const float* W      = (const float*)d_in[2];
  const float* T      = (const float*)d_in[3];
  const float* K      = (const float*)d_in[4];
  float* ws = (float*)d_ws;

  crf_prep <<<1,       256, 0, stream>>>(W, T, K, ws);
  crf_batch<<<B_BATCH, 256, 0, stream>>>(X, labels, T, ws);
  crf_final<<<1,       256, 0, stream>>>(ws, (float*)d_out);
}